// stnFFD_4320737100358
// MI455X (gfx1250) — compile-verified
//
#include <hip/hip_runtime.h>
#include <math.h>

typedef __attribute__((ext_vector_type(2))) float v2f;
typedef __attribute__((ext_vector_type(8))) float v8f;
typedef __attribute__((ext_vector_type(4))) float f4v;

#define IMG 120
#define TT  1024
#define NP  1024
#define NB  8
#define KFC 6760   // 10*26*26

// ---------------- conv1: (8,3,120,120) -> conv7x7 -> pool2 -> relu -> (8,8,57,57)
__global__ void conv1_kernel(const float* __restrict__ x, const float* __restrict__ w,
                             const float* __restrict__ b, float* __restrict__ out1) {
  int idx = blockIdx.x * blockDim.x + threadIdx.x;
  if (idx >= NB * 8 * 57 * 57) return;
  int pw = idx % 57; int t = idx / 57;
  int ph = t % 57;   t /= 57;
  int c  = t % 8;    int n = t / 8;
  float s00 = 0.f, s01 = 0.f, s10 = 0.f, s11 = 0.f;
  const float* wp = w + c * 3 * 49;
  for (int ci = 0; ci < 3; ++ci) {
    const float* xp  = x + ((n * 3 + ci) * IMG + 2 * ph) * IMG + 2 * pw;
    const float* wpc = wp + ci * 49;
    #pragma unroll
    for (int kh = 0; kh < 7; ++kh) {
      const float* xr0 = xp + kh * IMG;
      const float* xr1 = xr0 + IMG;
      #pragma unroll
      for (int kw = 0; kw < 7; ++kw) {
        float wv = wpc[kh * 7 + kw];
        s00 = fmaf(wv, xr0[kw],     s00);
        s01 = fmaf(wv, xr0[kw + 1], s01);
        s10 = fmaf(wv, xr1[kw],     s10);
        s11 = fmaf(wv, xr1[kw + 1], s11);
      }
    }
  }
  float m = fmaxf(fmaxf(s00, s01), fmaxf(s10, s11)) + b[c];
  out1[idx] = fmaxf(m, 0.f);
}

// ---------------- zero padding rows 8..15 of h (so WMMA A-loads need no predication)
__global__ void zero_hpad_kernel(float* __restrict__ p, int n) {
  int i = blockIdx.x * blockDim.x + threadIdx.x;
  if (i < n) p[i] = 0.f;
}

// ---------------- conv2: (8,8,57,57) -> conv5x5 -> pool2 -> relu -> h rows 0..7 (16 x 6760)
__global__ void conv2_kernel(const float* __restrict__ in, const float* __restrict__ w,
                             const float* __restrict__ b, float* __restrict__ h) {
  int idx = blockIdx.x * blockDim.x + threadIdx.x;
  if (idx >= NB * 10 * 26 * 26) return;
  int pw = idx % 26; int t = idx / 26;
  int ph = t % 26;   t /= 26;
  int c  = t % 10;   int n = t / 10;
  float s00 = 0.f, s01 = 0.f, s10 = 0.f, s11 = 0.f;
  const float* wp = w + c * 8 * 25;
  for (int ci = 0; ci < 8; ++ci) {
    const float* ip  = in + ((n * 8 + ci) * 57 + 2 * ph) * 57 + 2 * pw;
    const float* wpc = wp + ci * 25;
    #pragma unroll
    for (int kh = 0; kh < 5; ++kh) {
      const float* xr0 = ip + kh * 57;
      const float* xr1 = xr0 + 57;
      #pragma unroll
      for (int kw = 0; kw < 5; ++kw) {
        float wv = wpc[kh * 5 + kw];
        s00 = fmaf(wv, xr0[kw],     s00);
        s01 = fmaf(wv, xr0[kw + 1], s01);
        s10 = fmaf(wv, xr1[kw],     s10);
        s11 = fmaf(wv, xr1[kw + 1], s11);
      }
    }
  }
  float m = fmaxf(fmaxf(s00, s01), fmaxf(s10, s11)) + b[c];
  h[n * KFC + c * 676 + ph * 26 + pw] = fmaxf(m, 0.f);
}

// ---------------- per-plane fill constant: bilinear sample at (59.5, 59.5)
__global__ void fillval_kernel(const float* __restrict__ x, float* __restrict__ fv) {
  int t = threadIdx.x;
  if (t < NB * 3) {
    const float* xb = x + (size_t)t * IMG * IMG;
    fv[t] = 0.25f * (xb[59 * IMG + 59] + xb[59 * IMG + 60] +
                     xb[60 * IMG + 59] + xb[60 * IMG + 60]);
  }
}

// ---------------- FC1 (fp32 WMMA 16x16x4) + FC2 + Rodrigues -> T_theta (8x3x4)
// 1 block, 64 threads = 2 waves. Wave w computes N-tile w (cols 16w..16w+15).
// A = h (16 x 6760, rows 8..15 zero), B[k][n] = w_fc1[n][k].
__global__ void fc_wmma_kernel(const float* __restrict__ h,
                               const float* __restrict__ w1, const float* __restrict__ b1,
                               const float* __restrict__ w2, const float* __restrict__ b2,
                               float* __restrict__ Tout) {
  __shared__ float hfc[8][32];
  __shared__ float th[8][7];
  const int tid  = threadIdx.x;
  const int wave = tid >> 5;
  const int lane = tid & 31;
  const int half = lane >> 4;      // K sub-block: lanes 0-15 -> K{0,1}, 16-31 -> K{2,3}
  const int r    = lane & 15;      // A row (lanes<16) / B column
  const int ncol = wave * 16 + r;  // output column in [0,32)

  const float* ha = h  + r * KFC   + half * 2;  // A[r][k + 2*half + j]
  const float* wb = w1 + ncol * KFC + half * 2; // B[k + 2*half + j][ncol] = w1[ncol][...]

  v8f acc = {};
  for (int k = 0; k < KFC; k += 4) {
    v2f a, bb;
    a.x  = ha[k]; a.y  = ha[k + 1];
    bb.x = wb[k]; bb.y = wb[k + 1];
    // D = A(16x4, f32) * B(4x16, f32) + C  -> v_wmma_f32_16x16x4_f32
    acc = __builtin_amdgcn_wmma_f32_16x16x4_f32(
        false, a, false, bb, (short)0, acc, false, false);
  }
  // C/D layout: VGPR j -> M=j (lanes 0-15), M=j+8 (lanes 16-31). Rows 8..15 are padding.
  if (half == 0) {
    float bn = b1[ncol];
    #pragma unroll
    for (int j = 0; j < 8; ++j)
      hfc[j][ncol] = fmaxf(acc[j] + bn, 0.f);
  }
  __syncthreads();

  // FC2: theta[m][o] = hfc[m][:] . w2[o][:] + b2[o]
  if (tid < 56) {
    int m = tid / 7, o = tid % 7;
    float s = b2[o];
    #pragma unroll
    for (int i = 0; i < 32; ++i) s = fmaf(hfc[m][i], w2[o * 32 + i], s);
    th[m][o] = s;
  }
  __syncthreads();

  // Rodrigues + scale + translation -> T_theta row-major 3x4 per batch
  if (tid < NB) {
    float vx = th[tid][1], vy = th[tid][2], vz = th[tid][3];
    float ang = sqrtf(vx * vx + vy * vy + vz * vz + 1e-8f);
    float inv = 1.f / ang;
    float kx = vx * inv, ky = vy * inv, kz = vz * inv;
    float s = sinf(ang), c = 1.f - cosf(ang);
    float R[3][3];
    R[0][0] = 1.f + c * (-(ky * ky + kz * kz));
    R[0][1] = s * (-kz) + c * (kx * ky);
    R[0][2] = s * ( ky) + c * (kx * kz);
    R[1][0] = s * ( kz) + c * (kx * ky);
    R[1][1] = 1.f + c * (-(kx * kx + kz * kz));
    R[1][2] = s * (-kx) + c * (ky * kz);
    R[2][0] = s * (-ky) + c * (kx * kz);
    R[2][1] = s * ( kx) + c * (ky * kz);
    R[2][2] = 1.f + c * (-(kx * kx + ky * ky));
    float sc = fabsf(th[tid][0]);
    float* op = Tout + tid * 12;
    #pragma unroll
    for (int rr = 0; rr < 3; ++rr) {
      op[rr * 4 + 0] = R[rr][0] * sc;
      op[rr * 4 + 1] = R[rr][1] * sc;
      op[rr * 4 + 2] = R[rr][2] * sc;
      op[rr * 4 + 3] = th[tid][4 + rr];
    }
  }
}

// ---------------- streaming fill of aligned_x with per-plane constant (float4)
__global__ void fill_kernel(const float* __restrict__ fv, float* __restrict__ aligned) {
  size_t i = (size_t)blockIdx.x * blockDim.x + threadIdx.x;   // one float4 each
  int plane = (int)(i >> 18);                                 // 1024*1024/4 = 2^18
  float v = fv[plane];
  f4v val = {v, v, v, v};
  ((f4v*)aligned)[i] = val;
}

// ---------------- scatter: project P points per batch, bilinear sample, overwrite
__device__ __forceinline__ float gsamp(const float* __restrict__ xb, float iyf, float ixf) {
  bool valid = (ixf >= 0.f) && (ixf < (float)IMG) && (iyf >= 0.f) && (iyf < (float)IMG);
  int ii = (int)fminf(fmaxf(ixf, 0.f), (float)(IMG - 1));
  int jj = (int)fminf(fmaxf(iyf, 0.f), (float)(IMG - 1));
  float v = xb[jj * IMG + ii];
  return valid ? v : 0.f;
}

__global__ void scatter_kernel(const float* __restrict__ x, const float* __restrict__ uv,
                               const float* __restrict__ xyz, const float* __restrict__ Tm,
                               float* __restrict__ aligned) {
  int idx = blockIdx.x * blockDim.x + threadIdx.x;
  if (idx >= NB * NP) return;
  int k = idx & (NP - 1);
  int m = idx >> 10;
  float X = xyz[k * 3 + 0], Y = xyz[k * 3 + 1], Z = xyz[k * 3 + 2];
  const float* t = Tm + m * 12;
  float p0 = t[0] * X + t[1] * Y + t[2]  * Z + t[3];
  float p1 = t[4] * X + t[5] * Y + t[6]  * Z + t[7];
  float xs = p0 * (2.f / IMG) - 1.f;
  float ys = -(p1 * (2.f / IMG) - 1.f);
  int px = (int)floorf(uv[2 * k + 0] * (float)TT);
  int py = (int)floorf(uv[2 * k + 1] * (float)TT);
  float ix = ((xs + 1.f) * IMG - 1.f) * 0.5f;
  float iy = ((ys + 1.f) * IMG - 1.f) * 0.5f;
  float ix0 = floorf(ix), iy0 = floorf(iy);
  float wx1 = ix - ix0, wx0 = 1.f - wx1;
  float wy1 = iy - iy0, wy0 = 1.f - wy1;
  #pragma unroll
  for (int c = 0; c < 3; ++c) {
    const float* xb = x + (size_t)(m * 3 + c) * IMG * IMG;
    float v = gsamp(xb, iy0,       ix0)       * (wy0 * wx0)
            + gsamp(xb, iy0,       ix0 + 1.f) * (wy0 * wx1)
            + gsamp(xb, iy0 + 1.f, ix0)       * (wy1 * wx0)
            + gsamp(xb, iy0 + 1.f, ix0 + 1.f) * (wy1 * wx1);
    aligned[((size_t)(m * 3 + c) * TT + px) * TT + py] = v;
  }
}

extern "C" void kernel_launch(void* const* d_in, const int* in_sizes, int n_in,
                              void* d_out, int out_size, void* d_ws, size_t ws_size,
                              hipStream_t stream) {
  const float* x   = (const float*)d_in[0];
  const float* uv  = (const float*)d_in[1];
  const float* xyz = (const float*)d_in[2];
  const float* w1c = (const float*)d_in[3];
  const float* b1c = (const float*)d_in[4];
  const float* w2c = (const float*)d_in[5];
  const float* b2c = (const float*)d_in[6];
  const float* wf1 = (const float*)d_in[7];
  const float* bf1 = (const float*)d_in[8];
  const float* wf2 = (const float*)d_in[9];
  const float* bf2 = (const float*)d_in[10];
  float* out = (float*)d_out;
  float* ws  = (float*)d_ws;

  float* out1 = ws;                        // 8*8*57*57 = 207936 floats
  float* h    = ws + 207936;               // 16*6760   = 108160 floats (rows 8..15 zero)
  float* fv   = ws + 207936 + 108160;      // 24 floats

  // stage 1: conv1 + pool + relu
  conv1_kernel<<<(NB * 8 * 57 * 57 + 255) / 256, 256, 0, stream>>>(x, w1c, b1c, out1);
  // zero WMMA padding rows of h
  zero_hpad_kernel<<<(8 * KFC + 255) / 256, 256, 0, stream>>>(h + 8 * KFC, 8 * KFC);
  // stage 2: conv2 + pool + relu -> flattened h
  conv2_kernel<<<(NB * 10 * 26 * 26 + 255) / 256, 256, 0, stream>>>(out1, w2c, b2c, h);
  // per-plane fill constants (grid==0 => bilinear at image center)
  fillval_kernel<<<1, 32, 0, stream>>>(x, fv);
  // FC1 (fp32 WMMA) + FC2 + Rodrigues -> T_theta at out[0:96]
  fc_wmma_kernel<<<1, 64, 0, stream>>>(h, wf1, bf1, wf2, bf2, out);
  // bandwidth stage: fill 100MB aligned_x with per-plane constants
  fill_kernel<<<(NB * 3 * TT * TT / 4) / 256, 256, 0, stream>>>(fv, out + 96);
  // overwrite the 8*1024 scattered sample points
  scatter_kernel<<<(NB * NP + 255) / 256, 256, 0, stream>>>(x, uv, xyz, out, out + 96);
}